// RGCN_81028853006654
// MI455X (gfx1250) — compile-verified
//
#include <hip/hip_runtime.h>

typedef __attribute__((ext_vector_type(2))) float v2f;
typedef __attribute__((ext_vector_type(8))) float v8f;

#define NREL 8
#define DF 128  // input feature dim of every layer

// ---------------------------------------------------------------- utilities
__global__ void k_zero4(float4* __restrict__ p, long long n4) {
  long long i = (long long)blockIdx.x * blockDim.x + threadIdx.x;
  if (i < n4) p[i] = make_float4(0.f, 0.f, 0.f, 0.f);
}

__global__ void k_count(const int* __restrict__ dst, const int* __restrict__ et,
                        float* __restrict__ cnt, int E) {
  int e = blockIdx.x * blockDim.x + threadIdx.x;
  if (e < E) unsafeAtomicAdd(&cnt[(long long)dst[e] * NREL + et[e]], 1.0f);
}

__global__ void k_invert(float* __restrict__ cnt, int n) {
  int i = blockIdx.x * blockDim.x + threadIdx.x;
  if (i < n) cnt[i] = 1.0f / fmaxf(cnt[i], 1.0f);
}

// 128 threads per edge: coalesced read of h[src,:], f32 atomic scatter into S[dst,r,:]
__global__ void k_scatter(const float* __restrict__ h, const int* __restrict__ src,
                          const int* __restrict__ dst, const int* __restrict__ et,
                          float* __restrict__ S, int E) {
  long long t = (long long)blockIdx.x * blockDim.x + threadIdx.x;
  int e = (int)(t >> 7);
  int f = (int)(t & 127);
  if (e < E) {
    float v = h[(long long)src[e] * DF + f];
    unsafeAtomicAdd(&S[((long long)dst[e] * NREL + et[e]) * DF + f], v);
  }
}

// ------------------------------------------------------------- WMMA GEMM
// out[n, :] = sum_r (S[n,r,:] * inv[n,r]) @ W_r  +  h[n,:] @ root  +  b
// One block = 16 node rows. Wave w owns 16x16 output tile at columns [16w,16w+16).
// K loop = 8 relation chunks of 128 + 1 root chunk of 128, in K=4 WMMA steps.
template <int OUTD, bool RELU>
__global__ void k_gemm(const float* __restrict__ S, const float* __restrict__ invc,
                       const float* __restrict__ hin,
                       const float* __restrict__ W,     // [NREL, DF, OUTD]
                       const float* __restrict__ root,  // [DF, OUTD]
                       const float* __restrict__ bias,  // [OUTD]
                       float* __restrict__ out, int n) {
  __shared__ float lds[16 * 130];  // pad 128 -> 130 to avoid bank conflicts
  const int tid  = threadIdx.x;
  const int lane = tid & 31;
  const int wv   = tid >> 5;
  const int half = lane >> 4;  // 0: K+0/K+1, rows M+0..7 ; 1: K+2/K+3, rows M+8..15
  const int l15  = lane & 15;
  const int nbase = blockIdx.x * 16;
  const int col  = wv * 16 + l15;
  const int T    = blockDim.x;

  v8f acc = {};

  for (int chunk = 0; chunk < NREL + 1; ++chunk) {
    // stage scaled 16x128 A-tile
    for (int idx = tid; idx < 16 * DF; idx += T) {
      int row = idx >> 7, c = idx & 127;
      int node = nbase + row;
      float v = 0.f;
      if (node < n) {
        if (chunk < NREL) {
          long long seg = (long long)node * NREL + chunk;
          v = S[seg * DF + c] * invc[seg];
        } else {
          v = hin[(long long)node * DF + c];
        }
      }
      lds[row * 130 + c] = v;
    }
    __syncthreads();

    const float* Wsel = (chunk < NREL) ? (W + (long long)chunk * DF * OUTD) : root;
#pragma unroll 4
    for (int kk = 0; kk < DF; kk += 4) {
      int k0 = kk + half * 2;
      v2f a, b;
      a.x = lds[l15 * 130 + k0];
      a.y = lds[l15 * 130 + k0 + 1];
      b.x = Wsel[(long long)k0 * OUTD + col];
      b.y = Wsel[(long long)(k0 + 1) * OUTD + col];
      acc = __builtin_amdgcn_wmma_f32_16x16x4_f32(false, a, false, b, (short)0,
                                                  acc, false, false);
    }
    __syncthreads();
  }

  float bv = bias[col];
#pragma unroll
  for (int i = 0; i < 8; ++i) {
    int row = nbase + i + half * 8;
    float v = acc[i] + bv;
    if (RELU) v = fmaxf(v, 0.f);
    if (row < n) out[(long long)row * OUTD + col] = v;
  }
}

// ---------------------------------------------------------------- launcher
extern "C" void kernel_launch(void* const* d_in, const int* in_sizes, int n_in,
                              void* d_out, int out_size, void* d_ws, size_t ws_size,
                              hipStream_t stream) {
  const float* x     = (const float*)d_in[0];
  const int*   ei    = (const int*)d_in[1];  // [2, E] row-major: src then dst
  const int*   et    = (const int*)d_in[2];
  const float* W1    = (const float*)d_in[3];
  const float* root1 = (const float*)d_in[4];
  const float* b1    = (const float*)d_in[5];
  const float* W2    = (const float*)d_in[6];
  const float* root2 = (const float*)d_in[7];
  const float* b2    = (const float*)d_in[8];
  const float* W3    = (const float*)d_in[9];
  const float* root3 = (const float*)d_in[10];
  const float* b3    = (const float*)d_in[11];

  const int E = in_sizes[2];
  const int N = in_sizes[0] / DF;
  const int* src = ei;
  const int* dst = ei + E;

  float* ws  = (float*)d_ws;
  float* S   = ws;                                    // N*R*DF floats (204.8 MB)
  float* inv = S + (long long)N * NREL * DF;          // N*R floats
  float* h1  = inv + (long long)N * NREL;             // N*DF floats
  float* h2  = h1 + (long long)N * DF;                // N*DF floats
  float* outp = (float*)d_out;

  // per-(node,relation) counts -> 1/max(cnt,1)  (layer-invariant)
  {
    long long n4 = ((long long)N * NREL) / 4;
    k_zero4<<<(unsigned)((n4 + 255) / 256), 256, 0, stream>>>((float4*)inv, n4);
    k_count<<<(E + 255) / 256, 256, 0, stream>>>(dst, et, inv, E);
    int nr = N * NREL;
    k_invert<<<(nr + 255) / 256, 256, 0, stream>>>(inv, nr);
  }

  long long s4 = ((long long)N * NREL * DF) / 4;
  unsigned zb = (unsigned)((s4 + 255) / 256);
  unsigned sb = (unsigned)((((long long)E << 7) + 255) / 256);
  unsigned gb = (unsigned)((N + 15) / 16);

  // ---- layer 1: x -> h1 (O=128, ReLU)
  k_zero4<<<zb, 256, 0, stream>>>((float4*)S, s4);
  k_scatter<<<sb, 256, 0, stream>>>(x, src, dst, et, S, E);
  k_gemm<128, true><<<gb, 256, 0, stream>>>(S, inv, x, W1, root1, b1, h1, N);

  // ---- layer 2: h1 -> h2 (O=128, ReLU)
  k_zero4<<<zb, 256, 0, stream>>>((float4*)S, s4);
  k_scatter<<<sb, 256, 0, stream>>>(h1, src, dst, et, S, E);
  k_gemm<128, true><<<gb, 256, 0, stream>>>(S, inv, h1, W2, root2, b2, h2, N);

  // ---- layer 3: h2 -> out (O=64, no ReLU)
  k_zero4<<<zb, 256, 0, stream>>>((float4*)S, s4);
  k_scatter<<<sb, 256, 0, stream>>>(h2, src, dst, et, S, E);
  k_gemm<64, false><<<gb, 128, 0, stream>>>(S, inv, h2, W3, root3, b3, outp, N);
}